// GATLayer_72164040508142
// MI455X (gfx1250) — compile-verified
//
#include <hip/hip_runtime.h>
#include <hip/hip_bf16.h>
#include <stdint.h>

typedef __attribute__((ext_vector_type(2))) float v2f;
typedef __attribute__((ext_vector_type(8))) float v8f;

#define LEAK  0.01f
#define O_DIM 128
#define I_DIM 128
#define H_DIM 4
#define A_PAD 132   // 128 + 4-word pad: lanes hit banks 4l..4l+3, conflict-free

// ---------------------------------------------------------------------------
// Kernel 1: x[h] = input @ W[h]^T via V_WMMA_F32_16X16X4_F32.
// One block = one 16-row node strip; its 16x128 A panel is staged into LDS
// once with CDNA5 async copies (global_load_async_to_lds_b128 / ASYNCcnt),
// then 8 waves each own one 16-col o-tile and accumulate all 4 heads.
//   A 16x4 f32 frag: lanes 0-15 -> K = k,k+1 ; lanes 16-31 -> K = k+2,k+3
//   B 4x16 f32 frag: N striped across lanes (lane&15), K split the same way
//   C 16x16 f32:     VGPR r, lanes 0-15 -> M=r ; lanes 16-31 -> M=r+8
// ---------------------------------------------------------------------------
__global__ __launch_bounds__(256) void gat_gemm_wmma(
    const float* __restrict__ in, const float* __restrict__ W,
    float* __restrict__ x, float* __restrict__ out, int N)
{
    __shared__ float sA[16 * A_PAD];
    const int n0 = blockIdx.x * 16;

    // ---- async stage: 16 rows x 128 floats = 512 x 16B chunks, 2 per thread
    {
        const unsigned lds0 = (unsigned)(uintptr_t)(&sA[0]);
        #pragma unroll
        for (int j = 0; j < 2; ++j) {
            const int f   = (int)threadIdx.x + j * 256;   // 0..511
            const int row = f >> 5;
            const int col = (f & 31) * 4;
            const float* g = in + (size_t)(n0 + row) * I_DIM + col;
            const unsigned loff = lds0 + (unsigned)(row * A_PAD + col) * 4u;
            asm volatile("global_load_async_to_lds_b128 %0, %1, off"
                         :: "v"(loff), "v"(g) : "memory");
        }
        asm volatile("s_wait_asynccnt 0x0" ::: "memory");
        __syncthreads();
    }

    const int wave = threadIdx.x >> 5;
    const int o0   = wave * 16;
    const int lane = threadIdx.x & 31;
    const int half = lane >> 4;
    const int l    = lane & 15;

    const float* sRow = &sA[l * A_PAD + 2 * half];
    const size_t wStride = (size_t)O_DIM * I_DIM;
    const float* bP0 = W + (size_t)(o0 + l) * I_DIM + 2 * half;
    const float* bP1 = bP0 + wStride;
    const float* bP2 = bP1 + wStride;
    const float* bP3 = bP2 + wStride;

    v8f acc0 = {}; v8f acc1 = {}; v8f acc2 = {}; v8f acc3 = {};

    #pragma unroll 4
    for (int k = 0; k < I_DIM; k += 4) {
        v2f av; av.x = sRow[k]; av.y = sRow[k + 1];
        v2f b0; b0.x = bP0[k];  b0.y = bP0[k + 1];
        v2f b1; b1.x = bP1[k];  b1.y = bP1[k + 1];
        v2f b2; b2.x = bP2[k];  b2.y = bP2[k + 1];
        v2f b3; b3.x = bP3[k];  b3.y = bP3[k + 1];
        acc0 = __builtin_amdgcn_wmma_f32_16x16x4_f32(false, av, false, b0, (short)0, acc0, false, false);
        acc1 = __builtin_amdgcn_wmma_f32_16x16x4_f32(false, av, false, b1, (short)0, acc1, false, false);
        acc2 = __builtin_amdgcn_wmma_f32_16x16x4_f32(false, av, false, b2, (short)0, acc2, false, false);
        acc3 = __builtin_amdgcn_wmma_f32_16x16x4_f32(false, av, false, b3, (short)0, acc3, false, false);
    }

    const size_t hm = (size_t)N * O_DIM;
    #pragma unroll
    for (int r = 0; r < 8; ++r) {
        const int row = n0 + 8 * half + r;
        const int col = o0 + l;
        const size_t base = (size_t)row * O_DIM + col;
        const float v0 = acc0[r], v1 = acc1[r], v2 = acc2[r], v3 = acc3[r];
        x[base]          = v0;
        x[base + hm]     = v1;
        x[base + 2 * hm] = v2;
        x[base + 3 * hm] = v3;
        out[base] = (v0 + v1 + v2 + v3) * 0.25f;
    }
}

// ---------------------------------------------------------------------------
// Kernel 2: s_src[h,n] = dot(x[h,n,:], a_src[h]),  s_dst likewise.
// One wave per (h,n); coalesced 128-wide dot + shuffle reduction.
// ---------------------------------------------------------------------------
__global__ __launch_bounds__(256) void gat_node_scores(
    const float* __restrict__ x, const float* __restrict__ a,
    float* __restrict__ s_src, float* __restrict__ s_dst, int N)
{
    const int warp = (int)((blockIdx.x * blockDim.x + threadIdx.x) >> 5);
    const int lane = threadIdx.x & 31;
    if (warp >= H_DIM * N) return;
    const int h = warp / N;
    const int n = warp % N;

    const float* xr = x + ((size_t)h * N + n) * O_DIM;
    const float* as = a + (size_t)h * 2 * O_DIM;
    const float* ad = as + O_DIM;

    float ss = 0.f, sd = 0.f;
    #pragma unroll
    for (int j = 0; j < 4; ++j) {
        const int o = lane + 32 * j;
        const float v = xr[o];
        ss = fmaf(v, as[o], ss);
        sd = fmaf(v, ad[o], sd);
    }
    #pragma unroll
    for (int off = 16; off; off >>= 1) {
        ss += __shfl_down(ss, off);
        sd += __shfl_down(sd, off);
    }
    if (lane == 0) { s_src[warp] = ss; s_dst[warp] = sd; }
}

// Order-preserving float <-> uint map for atomicMax on floats of any sign.
__device__ __forceinline__ unsigned fenc(float f) {
    unsigned u = __float_as_uint(f);
    return (u & 0x80000000u) ? ~u : (u | 0x80000000u);
}
__device__ __forceinline__ float fdec(unsigned e) {
    return __uint_as_float((e & 0x80000000u) ? (e & 0x7fffffffu) : ~e);
}

__global__ void gat_init(unsigned* __restrict__ maxEnc, float* __restrict__ denom)
{
    if (threadIdx.x < H_DIM) { maxEnc[threadIdx.x] = 0u; denom[threadIdx.x] = 0.f; }
}

// ---------------------------------------------------------------------------
// Kernel 3: per-edge leaky-relu scores + per-head global max reduction.
// ---------------------------------------------------------------------------
__global__ __launch_bounds__(256) void gat_edge_scores(
    const int* __restrict__ edges, const float* __restrict__ s_src,
    const float* __restrict__ s_dst, float* __restrict__ scores,
    unsigned* __restrict__ maxEnc, int N, int E)
{
    __shared__ float red[H_DIM][8];
    const int e    = blockIdx.x * blockDim.x + threadIdx.x;
    const int lane = threadIdx.x & 31;
    const int warp = threadIdx.x >> 5;

    float sc[H_DIM];
    #pragma unroll
    for (int h = 0; h < H_DIM; ++h) sc[h] = -3.4e38f;

    if (e < E) {
        const int src = edges[3 * e + 0];
        const int dst = edges[3 * e + 2];
        #pragma unroll
        for (int h = 0; h < H_DIM; ++h) {
            float s = s_src[h * N + src] + s_dst[h * N + dst];
            s = (s > 0.f) ? s : LEAK * s;
            scores[(size_t)h * E + e] = s;
            sc[h] = s;
        }
    }
    #pragma unroll
    for (int h = 0; h < H_DIM; ++h) {
        float m = sc[h];
        #pragma unroll
        for (int off = 16; off; off >>= 1) m = fmaxf(m, __shfl_down(m, off));
        if (lane == 0) red[h][warp] = m;
    }
    __syncthreads();
    if (warp == 0) {
        #pragma unroll
        for (int h = 0; h < H_DIM; ++h) {
            float m = (lane < 8) ? red[h][lane] : -3.4e38f;
            #pragma unroll
            for (int off = 16; off; off >>= 1) m = fmaxf(m, __shfl_down(m, off));
            if (lane == 0) atomicMax(&maxEnc[h], fenc(m));
        }
    }
}

// ---------------------------------------------------------------------------
// Kernel 4: p = exp(score - max), in place; per-head denominator sum.
// ---------------------------------------------------------------------------
__global__ __launch_bounds__(256) void gat_edge_exp(
    float* __restrict__ scores, const unsigned* __restrict__ maxEnc,
    float* __restrict__ denom, int E)
{
    __shared__ float red[H_DIM][8];
    const int e    = blockIdx.x * blockDim.x + threadIdx.x;
    const int lane = threadIdx.x & 31;
    const int warp = threadIdx.x >> 5;

    float p[H_DIM];
    #pragma unroll
    for (int h = 0; h < H_DIM; ++h) p[h] = 0.f;

    if (e < E) {
        #pragma unroll
        for (int h = 0; h < H_DIM; ++h) {
            const float m = fdec(maxEnc[h]);
            const float v = __expf(scores[(size_t)h * E + e] - m);
            scores[(size_t)h * E + e] = v;
            p[h] = v;
        }
    }
    #pragma unroll
    for (int h = 0; h < H_DIM; ++h) {
        float s = p[h];
        #pragma unroll
        for (int off = 16; off; off >>= 1) s += __shfl_down(s, off);
        if (lane == 0) red[h][warp] = s;
    }
    __syncthreads();
    if (warp == 0) {
        #pragma unroll
        for (int h = 0; h < H_DIM; ++h) {
            float s = (lane < 8) ? red[h][lane] : 0.f;
            #pragma unroll
            for (int off = 16; off; off >>= 1) s += __shfl_down(s, off);
            if (lane == 0) atomicAdd(&denom[h], s);
        }
    }
}

// ---------------------------------------------------------------------------
// Kernel 5: warp per edge; sum heads in registers, one f32 atomic per
// (edge, channel). x (41 MB) and out (10 MB) are L2-resident.
// ---------------------------------------------------------------------------
__global__ __launch_bounds__(256) void gat_aggregate(
    const int* __restrict__ edges, const float* __restrict__ scores,
    const float* __restrict__ denom, const float* __restrict__ x,
    float* __restrict__ out, int N, int E)
{
    const int warpId = (int)((blockIdx.x * blockDim.x + threadIdx.x) >> 5);
    const int lane   = threadIdx.x & 31;
    if (warpId >= E) return;

    const int src = edges[3 * warpId + 0];
    const int dst = edges[3 * warpId + 2];

    float c[H_DIM];
    #pragma unroll
    for (int h = 0; h < H_DIM; ++h)
        c[h] = scores[(size_t)h * E + warpId] * 0.25f / denom[h];

    const size_t hm = (size_t)N * O_DIM;
    const float* xb = x + (size_t)dst * O_DIM;
    float* ob = out + (size_t)src * O_DIM;

    #pragma unroll
    for (int j = 0; j < 4; ++j) {
        const int o = lane + 32 * j;
        const float m = c[0] * xb[o] + c[1] * xb[hm + o]
                      + c[2] * xb[2 * hm + o] + c[3] * xb[3 * hm + o];
        atomicAdd(&ob[o], m);
    }
}

// ---------------------------------------------------------------------------
extern "C" void kernel_launch(void* const* d_in, const int* in_sizes, int n_in,
                              void* d_out, int out_size, void* d_ws, size_t ws_size,
                              hipStream_t stream)
{
    const float* in    = (const float*)d_in[0];
    const int*   edges = (const int*)d_in[1];
    const float* W     = (const float*)d_in[2];
    const float* a     = (const float*)d_in[3];
    float*       out   = (float*)d_out;

    const int N = in_sizes[0] / I_DIM;   // 20000
    const int E = in_sizes[1] / 3;       // 320000

    // Workspace layout (~46.7 MB)
    float*    x      = (float*)d_ws;
    float*    s_src  = x + (size_t)H_DIM * N * O_DIM;
    float*    s_dst  = s_src + (size_t)H_DIM * N;
    float*    scores = s_dst + (size_t)H_DIM * N;
    unsigned* maxEnc = (unsigned*)(scores + (size_t)H_DIM * E);
    float*    denom  = (float*)(maxEnc + H_DIM);

    gat_gemm_wmma<<<N / 16, 256, 0, stream>>>(in, W, x, out, N);
    gat_init<<<1, 32, 0, stream>>>(maxEnc, denom);
    const int nodeWarps = H_DIM * N;
    gat_node_scores<<<(nodeWarps + 7) / 8, 256, 0, stream>>>(x, a, s_src, s_dst, N);
    gat_edge_scores<<<(E + 255) / 256, 256, 0, stream>>>(edges, s_src, s_dst, scores, maxEnc, N, E);
    gat_edge_exp<<<(E + 255) / 256, 256, 0, stream>>>(scores, maxEnc, denom, E);
    gat_aggregate<<<(E + 7) / 8, 256, 0, stream>>>(edges, scores, denom, x, out, N, E);
}